// BidirectionalMambaMixer_50242527428689
// MI455X (gfx1250) — compile-verified
//
#include <hip/hip_runtime.h>
#include <hip/hip_bf16.h>

typedef float v2f __attribute__((ext_vector_type(2)));
typedef float v8f __attribute__((ext_vector_type(8)));

#define B_SZ   2
#define SEQ    1024
#define D_MODEL 1024
#define D_INNER 2048
#define D_STATE 16
#define DT_RANK 64
#define NX      (DT_RANK + 2 * D_STATE)   // 96
#define M_ROWS  (B_SZ * SEQ)              // 2048

// Workspace layout (float offsets), reused across the two directions.
#define OFF_XZ  0L                                  // [2048, 4096]  xi | z
#define OFF_U   (OFF_XZ + (long)M_ROWS * 4096)      // [2048, 2048]
#define OFF_DL  (OFF_U  + (long)M_ROWS * D_INNER)   // [2048, 2048]
#define OFF_Y   (OFF_DL + (long)M_ROWS * D_INNER)   // [2048, 2048]
#define OFF_XD  (OFF_Y  + (long)M_ROWS * D_INNER)   // [2048, 96]

__device__ __forceinline__ float sigmoidf_(float x) { return 1.0f / (1.0f + __expf(-x)); }

// ---------------------------------------------------------------------------
// NT GEMM on the f32 WMMA pipe:  C[M,N] = A[M,K] * W[N,K]^T
// MODE 0: plain store
// MODE 1: store softplus(acc + bias[col])
// MODE 2: store 0.5*acc            (FLIPC row remap optional)
// MODE 3: C += 0.5*acc             (FLIPC row remap optional)
// flipA (runtime): read A rows time-flipped within each batch of Lseq rows.
//
// Block = 256 threads = 8 waves (wave32). Wave tile = 16x64 (4 accumulators),
// block tile = 32x256. Wave id goes through readfirstlane so all tile
// coordinates are SGPRs: every guard is a scalar branch and EXEC stays all-1s
// around every v_wmma (ISA requirement). Partial-N subtiles clamp their W row
// base for loads and predicate only the store. K is processed in groups of 16
// (4 wmma k-steps): 20 b64 loads are issued, then 16 wmmas consume them.
// ---------------------------------------------------------------------------
template <int MODE, int FLIPC>
__global__ void gemm_nt(const float* __restrict__ A, int ldA,
                        const float* __restrict__ W, int ldW,
                        float* __restrict__ C, int ldC,
                        int M, int N, int K,
                        const float* __restrict__ bias,
                        int flipA, int Lseq) {
  const int tid  = threadIdx.x;
  const int w    = __builtin_amdgcn_readfirstlane(tid >> 5);  // uniform wave id
  const int lane = tid & 31;
  const int hl   = lane >> 4;          // half-wave selector (K phase)
  const int r    = lane & 15;          // row/col within 16

  const int m0 = blockIdx.x * 32  + (w >> 2) * 16;   // SGPR
  const int n0 = blockIdx.y * 256 + (w & 3) * 64;    // SGPR
  if (m0 >= M) return;
  if (n0 >= N) return;                                // wave-uniform scalar exit

  // Uniform subtile validity (SGPR compares -> scalar branches only).
  const bool v1 = (n0 + 16) < N;
  const bool v2 = (n0 + 32) < N;
  const bool v3 = (n0 + 48) < N;
  // Clamped load bases: invalid subtiles read row block 0 (in bounds), wmma
  // runs unconditionally, store is predicated by v1/v2/v3.
  const int nb1 = v1 ? (n0 + 16) : 0;
  const int nb2 = v2 ? (n0 + 32) : 0;
  const int nb3 = v3 ? (n0 + 48) : 0;

  // Per-lane A row (optionally time-flipped).
  int mrow = m0 + r;
  long arow = mrow;
  if (flipA) {
    int b = mrow / Lseq;
    int l = mrow - b * Lseq;
    arow = (long)b * Lseq + (Lseq - 1 - l);
  }
  const float* Ap  = A + arow * (long)ldA + 2 * hl;
  const float* Wp0 = W + (long)(n0  + r) * ldW + 2 * hl;
  const float* Wp1 = W + (long)(nb1 + r) * ldW + 2 * hl;
  const float* Wp2 = W + (long)(nb2 + r) * ldW + 2 * hl;
  const float* Wp3 = W + (long)(nb3 + r) * ldW + 2 * hl;

  v8f acc0 = {}, acc1 = {}, acc2 = {}, acc3 = {};

  for (int k0 = 0; k0 < K; k0 += 16) {      // 4 wmma k-steps per group
    v2f a[4], b0[4], b1[4], b2[4], b3[4];
#pragma unroll
    for (int s = 0; s < 4; ++s) {
      const int k = k0 + 4 * s;
      a[s].x  = Ap[k];   a[s].y  = Ap[k + 1];
      b0[s].x = Wp0[k];  b0[s].y = Wp0[k + 1];
      b1[s].x = Wp1[k];  b1[s].y = Wp1[k + 1];
      b2[s].x = Wp2[k];  b2[s].y = Wp2[k + 1];
      b3[s].x = Wp3[k];  b3[s].y = Wp3[k + 1];
    }
#pragma unroll
    for (int s = 0; s < 4; ++s) {
      acc0 = __builtin_amdgcn_wmma_f32_16x16x4_f32(false, a[s], false, b0[s], (short)0, acc0, false, false);
      acc1 = __builtin_amdgcn_wmma_f32_16x16x4_f32(false, a[s], false, b1[s], (short)0, acc1, false, false);
      acc2 = __builtin_amdgcn_wmma_f32_16x16x4_f32(false, a[s], false, b2[s], (short)0, acc2, false, false);
      acc3 = __builtin_amdgcn_wmma_f32_16x16x4_f32(false, a[s], false, b3[s], (short)0, acc3, false, false);
    }
  }

  // Epilogue. C/D layout: VGPR jj -> row m0 + jj + 8*hl, lane -> col n0j + r.
#pragma unroll
  for (int j = 0; j < 4; ++j) {
    if (j == 1 && !v1) continue;   // uniform scalar branch
    if (j == 2 && !v2) continue;
    if (j == 3 && !v3) continue;
    const int col = n0 + 16 * j + r;
    v8f acc = (j == 0) ? acc0 : (j == 1) ? acc1 : (j == 2) ? acc2 : acc3;
#pragma unroll
    for (int jj = 0; jj < 8; ++jj) {
      const int row = m0 + jj + 8 * hl;
      long crow = row;
      if (FLIPC) {
        int b = row / Lseq;
        int l = row - b * Lseq;
        crow = (long)b * Lseq + (Lseq - 1 - l);
      }
      float* p = C + crow * (long)ldC + col;
      const float v = acc[jj];
      if (MODE == 0) {
        *p = v;
      } else if (MODE == 1) {
        float t = v + bias[col];
        *p = (t > 20.0f) ? t : log1pf(__expf(t));   // softplus
      } else if (MODE == 2) {
        *p = 0.5f * v;
      } else {
        *p += 0.5f * v;
      }
    }
  }
}

// ---------------------------------------------------------------------------
// Depthwise causal conv (width 4) + bias + SiLU.
// xi = xz[:, 0:2048] (ld 4096) -> u [2048, 2048]
// ---------------------------------------------------------------------------
__global__ void conv_silu(const float* __restrict__ xz,
                          const float* __restrict__ convW,
                          const float* __restrict__ convB,
                          float* __restrict__ u) {
  const long idx = (long)blockIdx.x * blockDim.x + threadIdx.x;  // b*2^21 + l*2^11 + c
  const int c = (int)(idx & (D_INNER - 1));
  const int l = (int)((idx >> 11) & (SEQ - 1));
  const int b = (int)(idx >> 21);
  float s = convB[c];
#pragma unroll
  for (int j = 0; j < 4; ++j) {
    int t = l + j - 3;
    if (t >= 0)
      s += xz[(((long)b * SEQ + t) * 4096) + c] * convW[c * 4 + j];
  }
  u[idx] = s * sigmoidf_(s);  // SiLU
}

// ---------------------------------------------------------------------------
// Selective scan: one thread per (batch, channel), h[16] in registers.
// B_t/C_t (32 floats) staged in LDS once per timestep per block.
// Fuses y = (scan_y + u*Dp) * silu(z).
// ---------------------------------------------------------------------------
__global__ void scan_kernel(const float* __restrict__ delta,
                            const float* __restrict__ u,
                            const float* __restrict__ xz,     // z = cols 2048..4095
                            const float* __restrict__ xdbl,   // [2048, 96]
                            const float* __restrict__ Alog,
                            const float* __restrict__ Dp,
                            float* __restrict__ y) {
  __shared__ float sBC[32];                 // [0:16)=B_t, [16:32)=C_t
  const int tid = threadIdx.x;
  const int d = blockIdx.x * blockDim.x + tid;   // channel
  const int b = blockIdx.y;

  float Areg[D_STATE];
  float h[D_STATE];
#pragma unroll
  for (int n = 0; n < D_STATE; ++n) {
    Areg[n] = -__expf(Alog[(long)d * D_STATE + n]);
    h[n] = 0.0f;
  }
  const float dpc = Dp[d];

  for (int t = 0; t < SEQ; ++t) {
    const long row = (long)b * SEQ + t;
    if (tid < 32) sBC[tid] = xdbl[row * NX + DT_RANK + tid];
    __syncthreads();

    const float dt = delta[row * D_INNER + d];
    const float uu = u[row * D_INNER + d];
    const float du = dt * uu;
    float yv = 0.0f;
#pragma unroll
    for (int n = 0; n < D_STATE; ++n) {
      h[n] = h[n] * __expf(dt * Areg[n]) + du * sBC[n];
      yv += h[n] * sBC[16 + n];
    }
    const float zz = xz[row * 4096 + D_INNER + d];
    y[row * D_INNER + d] = (yv + uu * dpc) * (zz * sigmoidf_(zz));
    __syncthreads();
  }
}

// ---------------------------------------------------------------------------
extern "C" void kernel_launch(void* const* d_in, const int* in_sizes, int n_in,
                              void* d_out, int out_size, void* d_ws, size_t ws_size,
                              hipStream_t stream) {
  const float* x = (const float*)d_in[0];
  float* out = (float*)d_out;
  float* ws = (float*)d_ws;

  const dim3 blk(256);

  for (int dir = 0; dir < 2; ++dir) {
    const int base = 1 + dir * 9;
    const float* inW   = (const float*)d_in[base + 0];
    const float* convW = (const float*)d_in[base + 1];
    const float* convB = (const float*)d_in[base + 2];
    const float* xpW   = (const float*)d_in[base + 3];
    const float* dtW   = (const float*)d_in[base + 4];
    const float* dtB   = (const float*)d_in[base + 5];
    const float* Alog  = (const float*)d_in[base + 6];
    const float* Dp    = (const float*)d_in[base + 7];
    const float* outW  = (const float*)d_in[base + 8];

    // 1) xz = x @ inW^T   [2048, 4096]; backward dir reads x time-flipped.
    gemm_nt<0, 0><<<dim3(M_ROWS / 32, 4096 / 256), blk, 0, stream>>>(
        x, D_MODEL, inW, D_MODEL, ws + OFF_XZ, 4096,
        M_ROWS, 2 * D_INNER, D_MODEL, nullptr, dir, SEQ);

    // 2) depthwise conv + SiLU -> u
    conv_silu<<<(B_SZ * SEQ * D_INNER) / 256, blk, 0, stream>>>(
        ws + OFF_XZ, convW, convB, ws + OFF_U);

    // 3) x_dbl = u @ xpW^T   [2048, 96]
    gemm_nt<0, 0><<<dim3(M_ROWS / 32, 1), blk, 0, stream>>>(
        ws + OFF_U, D_INNER, xpW, D_INNER, ws + OFF_XD, NX,
        M_ROWS, NX, D_INNER, nullptr, 0, SEQ);

    // 4) delta = softplus(dt @ dtW^T + dtB)   [2048, 2048]
    gemm_nt<1, 0><<<dim3(M_ROWS / 32, D_INNER / 256), blk, 0, stream>>>(
        ws + OFF_XD, NX, dtW, DT_RANK, ws + OFF_DL, D_INNER,
        M_ROWS, D_INNER, DT_RANK, dtB, 0, SEQ);

    // 5) selective scan + gating -> y   [2048, 2048]
    scan_kernel<<<dim3(D_INNER / 256, B_SZ), blk, 0, stream>>>(
        ws + OFF_DL, ws + OFF_U, ws + OFF_XZ, ws + OFF_XD, Alog, Dp, ws + OFF_Y);

    // 6) out  (+)= 0.5 * y @ outW^T; backward dir stores rows time-flipped.
    if (dir == 0) {
      gemm_nt<2, 0><<<dim3(M_ROWS / 32, D_MODEL / 256), blk, 0, stream>>>(
          ws + OFF_Y, D_INNER, outW, D_INNER, out, D_MODEL,
          M_ROWS, D_MODEL, D_INNER, nullptr, 0, SEQ);
    } else {
      gemm_nt<3, 1><<<dim3(M_ROWS / 32, D_MODEL / 256), blk, 0, stream>>>(
          ws + OFF_Y, D_INNER, outW, D_INNER, out, D_MODEL,
          M_ROWS, D_MODEL, D_INNER, nullptr, 0, SEQ);
    }
  }
}